// ShapegraphEncoder_17540646437245
// MI455X (gfx1250) — compile-verified
//
#include <hip/hip_runtime.h>
#include <hip/hip_bf16.h>

// ---------------- problem constants (match reference) ----------------
#define NN      20000          // nodes
#define EG      320000         // graph edges
#define BG      32             // graphs (num summary tokens, T==1)
#define OFF     32             // summary rows prepended
#define NT      20032          // total rows = OFF + NN
#define EE      380032         // EG + NN + NN + NT  (graph + s->n + n->s + self loops)
#define HID     256
#define HEADS   4
#define DH      64
#define NEG_SLOPE 0.2f

typedef float v2f __attribute__((ext_vector_type(2)));
typedef float v8f __attribute__((ext_vector_type(8)));

// ---------------- edge decode (edges are never materialized) ----------------
__device__ __forceinline__ void edge_decode(int i, const int* __restrict__ ei,
                                            const int* __restrict__ batch,
                                            int& s, int& d) {
    if (i < EG)             { s = ei[i] + OFF;        d = ei[EG + i] + OFF; }
    else if (i < EG + NN)   { int j = i - EG;         s = batch[j];  d = j + OFF; }
    else if (i < EG + 2*NN) { int j = i - EG - NN;    s = j + OFF;   d = batch[j]; }
    else                    { int j = i - EG - 2*NN;  s = j;         d = j; }
}

__device__ __forceinline__ void atomicMaxFloat(float* addr, float val) {
    int iv = __float_as_int(val);
    if (iv >= 0) atomicMax((int*)addr, iv);
    else         atomicMin((unsigned int*)addr, (unsigned int)iv);
}

// ---------------- f32 WMMA GEMM, 2x2 register-blocked --------------------
// C[M x 256] = A[M x K] * W[K x 256] (+bias). One wave computes a 32x32
// output block as four 16x16 WMMA tiles, reusing each A/B fragment twice:
//   per K-step (4): 2 A frags + 2 B frags -> 4 x V_WMMA_F32_16X16X4_F32.
// Fragment layouts (wave32):
//   A (16x4): lanes 0-15 M=lane, v0:K=k0+0 v1:K=k0+1 ; lanes 16-31 K=k0+2/3
//   B (4x16): lanes 0-15 N=lane, v0:K=k0+0 v1:K=k0+1 ; lanes 16-31 K=k0+2/3
//   C tile:   vgpr r, lanes 0-15 (M=r,N=l) ; lanes 16-31 (M=8+r,N=l)
// Requires M % 32 == 0 for full blocks (20000 / 20032 / 32 all qualify).
__global__ void gemm_wmma_f32_2x2(const float* __restrict__ A, int lda, int M,
                                  const float* __restrict__ W, int K,
                                  const float* __restrict__ bias,
                                  float* __restrict__ C, int ldc) {
    const int wave = (blockIdx.x * blockDim.x + threadIdx.x) >> 5;
    const int lane = threadIdx.x & 31;
    const int mt = wave >> 3;          // 8 N-superblocks (256/32)
    const int nt = wave & 7;
    const int m0 = mt * 32;
    if (m0 >= M) return;
    const int n0 = nt * 32;
    const int half = lane >> 4;        // 0: K lanes 0/1, 1: K lanes 2/3
    const int l    = lane & 15;

    const float* arow0 = A + (size_t)(m0 + l) * lda;
    const float* arow1 = A + (size_t)(m0 + 16 + l) * lda;
    const float* wcol0 = W + n0 + l;
    const float* wcol1 = W + n0 + 16 + l;

    v8f c00 = {}, c01 = {}, c10 = {}, c11 = {};
    for (int k0 = 0; k0 < K; k0 += 4) {
        const int ka = k0 + half * 2;
        v2f a0, a1, b0, b1;
        a0.x = arow0[ka];                  a0.y = arow0[ka + 1];
        a1.x = arow1[ka];                  a1.y = arow1[ka + 1];
        b0.x = wcol0[(size_t)ka * 256];    b0.y = wcol0[(size_t)(ka + 1) * 256];
        b1.x = wcol1[(size_t)ka * 256];    b1.y = wcol1[(size_t)(ka + 1) * 256];
        c00 = __builtin_amdgcn_wmma_f32_16x16x4_f32(false, a0, false, b0, (short)0, c00, false, false);
        c01 = __builtin_amdgcn_wmma_f32_16x16x4_f32(false, a0, false, b1, (short)0, c01, false, false);
        c10 = __builtin_amdgcn_wmma_f32_16x16x4_f32(false, a1, false, b0, (short)0, c10, false, false);
        c11 = __builtin_amdgcn_wmma_f32_16x16x4_f32(false, a1, false, b1, (short)0, c11, false, false);
    }
    const float bv0 = bias ? bias[n0 + l]      : 0.0f;
    const float bv1 = bias ? bias[n0 + 16 + l] : 0.0f;
#pragma unroll
    for (int r = 0; r < 8; ++r) {
        const int mm0 = m0 + half * 8 + r;
        const int mm1 = mm0 + 16;
        C[(size_t)mm0 * ldc + n0 + l]      = c00[r] + bv0;
        C[(size_t)mm0 * ldc + n0 + 16 + l] = c01[r] + bv1;
        C[(size_t)mm1 * ldc + n0 + l]      = c10[r] + bv0;
        C[(size_t)mm1 * ldc + n0 + 16 + l] = c11[r] + bv1;
    }
}

// ---------------- summary token broadcast into h[0:32] ----------------
__global__ void k_init_summary(const float* __restrict__ st, float* __restrict__ h) {
    int i = blockIdx.x * blockDim.x + threadIdx.x;
    if (i < OFF * HID) h[i] = st[i & (HID - 1)];   // T == 1: same row repeated
}

// ---------------- per-layer scratch init ----------------
__global__ void k_init_layer(float* __restrict__ accum, float* __restrict__ mmax,
                             float* __restrict__ ssum) {
    int i = blockIdx.x * blockDim.x + threadIdx.x;
    if (i < NT * HID) accum[i] = 0.0f;
    if (i < NT * HEADS) { mmax[i] = -__builtin_inff(); ssum[i] = 0.0f; }
}

// ---------------- attention logits: al_src/al_dst = sum(xp * a) over DH ------
__global__ void k_attn_logits(const float* __restrict__ xp,
                              const float* __restrict__ asrc,
                              const float* __restrict__ adst,
                              float* __restrict__ als, float* __restrict__ ald) {
    int t = blockIdx.x * blockDim.x + threadIdx.x;
    if (t >= NT * HEADS) return;
    const int n = t >> 2, hh = t & 3;
    const float* xrow = xp + (size_t)n * HID + hh * DH;
    const float* as = asrc + hh * DH;
    const float* ad = adst + hh * DH;
    float s1 = 0.f, s2 = 0.f;
#pragma unroll 8
    for (int d = 0; d < DH; ++d) { float x = xrow[d]; s1 += x * as[d]; s2 += x * ad[d]; }
    als[t] = s1; ald[t] = s2;
}

// ---------------- edge pass 1: segment max of leaky(e) over dst --------------
__global__ void k_edge_max(const int* __restrict__ ei, const int* __restrict__ batch,
                           const float* __restrict__ als, const float* __restrict__ ald,
                           float* __restrict__ mmax) {
    int t = blockIdx.x * blockDim.x + threadIdx.x;
    if (t >= EE * HEADS) return;
    const int e = t >> 2, hh = t & 3;
    int s, d; edge_decode(e, ei, batch, s, d);
    float v = als[s * HEADS + hh] + ald[d * HEADS + hh];
    v = v > 0.f ? v : NEG_SLOPE * v;
    atomicMaxFloat(&mmax[d * HEADS + hh], v);
}

// ---------------- edge pass 2: accumulate exp and unnormalized messages ------
// one wave32 per edge; lane l covers elements l + 32*j (coalesced), head = idx>>6
__global__ void k_edge_accum(const int* __restrict__ ei, const int* __restrict__ batch,
                             const float* __restrict__ als, const float* __restrict__ ald,
                             const float* __restrict__ mmax, const float* __restrict__ xp,
                             float* __restrict__ ssum, float* __restrict__ accum) {
    const int wave = (blockIdx.x * blockDim.x + threadIdx.x) >> 5;
    const int lane = threadIdx.x & 31;
    if (wave >= EE) return;
    int s, d; edge_decode(wave, ei, batch, s, d);
    float ex[HEADS];
#pragma unroll
    for (int hh = 0; hh < HEADS; ++hh) {
        float v = als[s * HEADS + hh] + ald[d * HEADS + hh];
        v = v > 0.f ? v : NEG_SLOPE * v;
        ex[hh] = __expf(v - mmax[d * HEADS + hh]);
    }
    if (lane < HEADS) atomicAdd(&ssum[d * HEADS + lane], ex[lane]);
    const float* xs = xp + (size_t)s * HID;
    float* ad = accum + (size_t)d * HID;
#pragma unroll
    for (int j = 0; j < 8; ++j) {
        const int idx = lane + 32 * j;
        atomicAdd(&ad[idx], ex[idx >> 6] * xs[idx]);
    }
}

// ---------------- node update: out = accum/s + bias; h = LN(h + out)*g + b ---
__global__ void k_node_update(const float* __restrict__ accum, const float* __restrict__ ssum,
                              const float* __restrict__ gbias, const float* __restrict__ g,
                              const float* __restrict__ b, float* __restrict__ h) {
    const int wave = (blockIdx.x * blockDim.x + threadIdx.x) >> 5;
    const int lane = threadIdx.x & 31;
    if (wave >= NT) return;
    const float* an = accum + (size_t)wave * HID;
    float* hn = h + (size_t)wave * HID;
    float v[8];
    float sum = 0.f;
#pragma unroll
    for (int j = 0; j < 8; ++j) {
        const int idx = lane + 32 * j;
        const float o = an[idx] / (ssum[wave * HEADS + (idx >> 6)] + 1e-16f) + gbias[idx];
        v[j] = hn[idx] + o;
        sum += v[j];
    }
#pragma unroll
    for (int off = 16; off; off >>= 1) sum += __shfl_xor(sum, off, 32);
    const float mu = sum * (1.0f / HID);
    float vs = 0.f;
#pragma unroll
    for (int j = 0; j < 8; ++j) { const float t = v[j] - mu; vs += t * t; }
#pragma unroll
    for (int off = 16; off; off >>= 1) vs += __shfl_xor(vs, off, 32);
    const float rstd = rsqrtf(vs * (1.0f / HID) + 1e-5f);
#pragma unroll
    for (int j = 0; j < 8; ++j) {
        const int idx = lane + 32 * j;
        hn[idx] = (v[j] - mu) * rstd * g[idx] + b[idx];
    }
}

// ---------------- host orchestration ----------------
extern "C" void kernel_launch(void* const* d_in, const int* in_sizes, int n_in,
                              void* d_out, int out_size, void* d_ws, size_t ws_size,
                              hipStream_t stream) {
    const float* x        = (const float*)d_in[0];
    const int*   ei       = (const int*)  d_in[1];
    const int*   batch    = (const int*)  d_in[2];
    /* d_in[3] = num_graphs (compile-time 32) */
    const float* W_in     = (const float*)d_in[4];
    const float* b_in     = (const float*)d_in[5];
    const float* st       = (const float*)d_in[6];
    const float* lin_w    = (const float*)d_in[7];
    const float* att_src  = (const float*)d_in[8];
    const float* att_dst  = (const float*)d_in[9];
    const float* gat_bias = (const float*)d_in[10];
    const float* ln_g     = (const float*)d_in[11];
    const float* ln_b     = (const float*)d_in[12];
    const float* W_out    = (const float*)d_in[13];
    const float* b_out    = (const float*)d_in[14];

    float* h     = (float*)d_ws;                 // NT*256
    float* xp    = h     + (size_t)NT * HID;     // NT*256
    float* accum = xp    + (size_t)NT * HID;     // NT*256
    float* als   = accum + (size_t)NT * HID;     // NT*4
    float* ald   = als   + (size_t)NT * HEADS;
    float* mmax  = ald   + (size_t)NT * HEADS;
    float* ssum  = mmax  + (size_t)NT * HEADS;

    // h[0:32] = summary tokens; h[32:] = x @ W_in + b_in  (WMMA f32, 2x2 tiles)
    k_init_summary<<<(OFF * HID + 255) / 256, 256, 0, stream>>>(st, h);
    {
        const int waves = (NN / 32) * 8;                            // 5000
        gemm_wmma_f32_2x2<<<(waves * 32 + 255) / 256, 256, 0, stream>>>(
            x, 128, NN, W_in, 128, b_in, h + (size_t)OFF * HID, HID);
    }

    const int gemm_waves = (NT / 32) * 8;                           // 5008
    for (int l = 0; l < 3; ++l) {
        gemm_wmma_f32_2x2<<<(gemm_waves * 32 + 255) / 256, 256, 0, stream>>>(
            h, HID, NT, lin_w + (size_t)l * HID * HID, HID, nullptr, xp, HID);
        k_attn_logits<<<(NT * HEADS + 255) / 256, 256, 0, stream>>>(
            xp, att_src + l * HEADS * DH, att_dst + l * HEADS * DH, als, ald);
        k_init_layer<<<((size_t)NT * HID + 255) / 256, 256, 0, stream>>>(accum, mmax, ssum);
        k_edge_max<<<((size_t)EE * HEADS + 255) / 256, 256, 0, stream>>>(ei, batch, als, ald, mmax);
        k_edge_accum<<<((size_t)EE * 32 + 255) / 256, 256, 0, stream>>>(
            ei, batch, als, ald, mmax, xp, ssum, accum);
        k_node_update<<<((size_t)NT * 32 + 255) / 256, 256, 0, stream>>>(
            accum, ssum, gat_bias + l * HID, ln_g + l * HID, ln_b + l * HID, h);
    }

    // out = h[:32] @ W_out + b_out -> (32, 1, 256) flat in d_out
    {
        const int waves = (OFF / 32) * 8;                           // 8
        gemm_wmma_f32_2x2<<<(waves * 32 + 255) / 256, 256, 0, stream>>>(
            h, HID, OFF, W_out, HID, b_out, (float*)d_out, 256);
    }
}